// SparNet_53867479826654
// MI455X (gfx1250) — compile-verified
//
#include <hip/hip_runtime.h>
#include <hip/hip_bf16.h>
#include <math.h>

typedef float v2f __attribute__((ext_vector_type(2)));
typedef float v8f __attribute__((ext_vector_type(8)));

#define SEQ   214
#define DIMS  80
#define DS1   128
#define DS3   128
#define VOCAB 95
#define MAXL2 80
#define NB    64

// ---------------------------------------------------------------------------
// Kernel 1: conv1 (5x5, no pad) -> maxpool 3x3 s2 ceil -> relu
// x: (64,1,28,860) -> h1: (64,8,12,428)
// ---------------------------------------------------------------------------
__global__ __launch_bounds__(256) void conv1_pool_kernel(
    const float* __restrict__ x, const float* __restrict__ w,
    const float* __restrict__ bias, float* __restrict__ h1)
{
    int idx = blockIdx.x * 256 + threadIdx.x;
    const int total = NB * 8 * 12 * 428;
    if (idx >= total) return;
    int wp = idx % 428; int t = idx / 428;
    int hp = t % 12;    t /= 12;
    int co = t % 8;     int b = t / 8;

    float wr[25];
#pragma unroll
    for (int i = 0; i < 25; ++i) wr[i] = w[co * 25 + i];
    float bb = bias[co];
    const float* xb = x + b * 28 * 860;

    float mx = -INFINITY;
    for (int ph = 0; ph < 3; ++ph) {
        int hc = 2 * hp + ph;
        if (hc >= 24) break;
        for (int pw = 0; pw < 3; ++pw) {
            int wc = 2 * wp + pw;
            if (wc >= 856) break;
            float acc = bb;
#pragma unroll
            for (int kh = 0; kh < 5; ++kh)
#pragma unroll
                for (int kw = 0; kw < 5; ++kw)
                    acc = fmaf(xb[(hc + kh) * 860 + wc + kw], wr[kh * 5 + kw], acc);
            mx = fmaxf(mx, acc);
        }
    }
    h1[idx] = fmaxf(mx, 0.f);
}

// ---------------------------------------------------------------------------
// Kernel 2: conv2 (3x3, pad W=1) -> maxpool 3x3 s2 ceil -> relu -> transpose
// h1: (64,8,12,428) -> enc0: (64,214,80)   [enc0[b][s][co*5+hp]]
// ---------------------------------------------------------------------------
__global__ __launch_bounds__(256) void conv2_pool_kernel(
    const float* __restrict__ h1, const float* __restrict__ w,
    const float* __restrict__ bias, float* __restrict__ enc0)
{
    int idx = blockIdx.x * 256 + threadIdx.x;
    const int total = NB * 16 * 5 * SEQ;
    if (idx >= total) return;
    int wp = idx % SEQ; int t = idx / SEQ;
    int hp = t % 5;     t /= 5;
    int co = t % 16;    int b = t / 16;

    float wr[72];
#pragma unroll
    for (int i = 0; i < 72; ++i) wr[i] = w[co * 72 + i];
    float bb = bias[co];
    const float* hb = h1 + b * 8 * 12 * 428;

    float mx = -INFINITY;
    for (int ph = 0; ph < 3; ++ph) {
        int hc = 2 * hp + ph;
        if (hc >= 10) break;
        for (int pw = 0; pw < 3; ++pw) {
            int wc = 2 * wp + pw;
            if (wc >= 428) break;
            float acc = bb;
#pragma unroll
            for (int ci = 0; ci < 8; ++ci)
#pragma unroll
                for (int kh = 0; kh < 3; ++kh) {
                    const float* rp = hb + ci * (12 * 428) + (hc + kh) * 428;
#pragma unroll
                    for (int kw = 0; kw < 3; ++kw) {
                        int c = wc - 1 + kw;
                        if (c >= 0 && c < 428)
                            acc = fmaf(rp[c], wr[ci * 9 + kh * 3 + kw], acc);
                    }
                }
            mx = fmaxf(mx, acc);
        }
    }
    enc0[b * (SEQ * DIMS) + wp * DIMS + co * 5 + hp] = fmaxf(mx, 0.f);
}

// ---------------------------------------------------------------------------
// WMMA fp32 GEMM helper types; fragment layouts per CDNA5 ISA 7.12.2:
//   A 16x4:  lanes 0-15 hold K0,K1 ; lanes 16-31 hold K2,K3 (VGPR0/1)
//   B 4x16:  VGPR0 = rows K0(l0-15)/K2(l16-31), VGPR1 = K1/K3
//   C/D 16x16: VGPR v -> M=v (lanes 0-15), M=v+8 (lanes 16-31), N=lane&15
// ---------------------------------------------------------------------------

// Kernel 3: encoder dense: relu(X@W1^T+b1) -> LN -> @W2^T+b2+pos -> LN
// enc0 (13696,80) -> enc (13696,80). 214 blocks x 128 threads (4 waves x 16 rows)
__global__ __launch_bounds__(128) void encoder_mlp_kernel(
    const float* __restrict__ enc0,
    const float* __restrict__ w1, const float* __restrict__ b1,
    const float* __restrict__ g1, const float* __restrict__ be1,
    const float* __restrict__ w2, const float* __restrict__ b2,
    const float* __restrict__ g2, const float* __restrict__ be2,
    float* __restrict__ enc)
{
    __shared__ float sX[64 * 80];    // input rows, later: GEMM2 output rows
    __shared__ float sT[64 * 128];   // GEMM1 output / LN'd hidden

    const int tid  = threadIdx.x;
    const int wid  = tid >> 5;
    const int lane = tid & 31;
    const int r    = lane & 15;
    const int koff = (lane >> 4) << 1;      // 0 or 2
    const int mloc = (lane >> 4) << 3;      // 0 or 8
    const int row0 = blockIdx.x * 64;

    for (int i = tid; i < 64 * 80; i += 128) sX[i] = enc0[row0 * 80 + i];
    __syncthreads();

    // ---- GEMM1: (64x80) @ (80x128), N tiles of 16, fp32 WMMA 16x16x4
    v8f acc[8];
#pragma unroll
    for (int nt = 0; nt < 8; ++nt)
#pragma unroll
        for (int e = 0; e < 8; ++e) acc[nt][e] = 0.f;

    const float* aRow = &sX[(wid * 16 + r) * 80];
    for (int kb = 0; kb < 80; kb += 4) {
        v2f a; a[0] = aRow[kb + koff]; a[1] = aRow[kb + koff + 1];
#pragma unroll
        for (int nt = 0; nt < 8; ++nt) {
            int n = nt * 16 + r;
            v2f bf; bf[0] = w1[n * 80 + kb + koff]; bf[1] = w1[n * 80 + kb + koff + 1];
            acc[nt] = __builtin_amdgcn_wmma_f32_16x16x4_f32(
                false, a, false, bf, (short)0, acc[nt], false, false);
        }
    }
#pragma unroll
    for (int nt = 0; nt < 8; ++nt) {
        int n = nt * 16 + r;
        float bias = b1[n];
#pragma unroll
        for (int v = 0; v < 8; ++v)
            sT[(wid * 16 + mloc + v) * 128 + n] = fmaxf(acc[nt][v] + bias, 0.f);
    }
    __syncthreads();

    // ---- LayerNorm over 128 (unbiased std, eps on std)
    if (tid < 64) {
        float m = 0.f;
        for (int k = 0; k < 128; ++k) m += sT[tid * 128 + k];
        m *= (1.f / 128.f);
        float v = 0.f;
        for (int k = 0; k < 128; ++k) { float d = sT[tid * 128 + k] - m; v = fmaf(d, d, v); }
        float sd = sqrtf(v / 127.f) + 1e-6f;
        float inv = 1.f / sd;
        for (int k = 0; k < 128; ++k)
            sT[tid * 128 + k] = g1[k] * (sT[tid * 128 + k] - m) * inv + be1[k];
    }
    __syncthreads();

    // ---- GEMM2: (64x128) @ (128x80), 5 N-tiles
    v8f acc2[5];
#pragma unroll
    for (int nt = 0; nt < 5; ++nt)
#pragma unroll
        for (int e = 0; e < 8; ++e) acc2[nt][e] = 0.f;

    const float* aRow2 = &sT[(wid * 16 + r) * 128];
    for (int kb = 0; kb < 128; kb += 4) {
        v2f a; a[0] = aRow2[kb + koff]; a[1] = aRow2[kb + koff + 1];
#pragma unroll
        for (int nt = 0; nt < 5; ++nt) {
            int n = nt * 16 + r;
            v2f bf; bf[0] = w2[n * 128 + kb + koff]; bf[1] = w2[n * 128 + kb + koff + 1];
            acc2[nt] = __builtin_amdgcn_wmma_f32_16x16x4_f32(
                false, a, false, bf, (short)0, acc2[nt], false, false);
        }
    }
    // add bias + positional encoding, stage into sX (input no longer needed)
#pragma unroll
    for (int nt = 0; nt < 5; ++nt) {
        int n = nt * 16 + r;
        float bias = b2[n];
        int i = n >> 1;
        float div = expf(-9.210340371976184f * (float)(2 * i) / 80.0f);
#pragma unroll
        for (int v = 0; v < 8; ++v) {
            int m = wid * 16 + mloc + v;
            int s = (row0 + m) % SEQ;
            float ang = (float)s * div;
            float pe = (n & 1) ? cosf(ang) : sinf(ang);
            sX[m * 80 + n] = acc2[nt][v] + bias + pe;
        }
    }
    __syncthreads();

    // ---- LayerNorm over 80, write enc
    if (tid < 64) {
        float m = 0.f;
        for (int k = 0; k < 80; ++k) m += sX[tid * 80 + k];
        m *= (1.f / 80.f);
        float v = 0.f;
        for (int k = 0; k < 80; ++k) { float d = sX[tid * 80 + k] - m; v = fmaf(d, d, v); }
        float inv = 1.f / (sqrtf(v / 79.f) + 1e-6f);
        for (int k = 0; k < 80; ++k)
            enc[(row0 + tid) * 80 + k] = g2[k] * (sX[tid * 80 + k] - m) * inv + be2[k];
    }
}

// ---------------------------------------------------------------------------
// Scan kernel helpers
// ---------------------------------------------------------------------------
__device__ __forceinline__ void softmax_shared(float* v, int n, float* red)
{
    int tid = threadIdx.x;
    float m = -INFINITY;
    for (int i = tid; i < n; i += 256) m = fmaxf(m, v[i]);
    red[tid] = m; __syncthreads();
    for (int s = 128; s > 0; s >>= 1) {
        if (tid < s) red[tid] = fmaxf(red[tid], red[tid + s]);
        __syncthreads();
    }
    m = red[0]; __syncthreads();
    float loc = 0.f;
    for (int i = tid; i < n; i += 256) { float e = expf(v[i] - m); v[i] = e; loc += e; }
    red[tid] = loc; __syncthreads();
    for (int s = 128; s > 0; s >>= 1) {
        if (tid < s) red[tid] += red[tid + s];
        __syncthreads();
    }
    float inv = 1.f / red[0]; __syncthreads();
    for (int i = tid; i < n; i += 256) v[i] *= inv;
    __syncthreads();
}

__device__ __forceinline__ void layernorm_shared(
    float* v, int n, const float* g, const float* be, float* red, float ddof)
{
    int tid = threadIdx.x;
    float loc = 0.f;
    for (int i = tid; i < n; i += 256) loc += v[i];
    red[tid] = loc; __syncthreads();
    for (int s = 128; s > 0; s >>= 1) {
        if (tid < s) red[tid] += red[tid + s];
        __syncthreads();
    }
    float m = red[0] / (float)n; __syncthreads();
    loc = 0.f;
    for (int i = tid; i < n; i += 256) { float d = v[i] - m; loc = fmaf(d, d, loc); }
    red[tid] = loc; __syncthreads();
    for (int s = 128; s > 0; s >>= 1) {
        if (tid < s) red[tid] += red[tid + s];
        __syncthreads();
    }
    float inv = 1.f / (sqrtf(red[0] / ddof) + 1e-6f); __syncthreads();
    for (int i = tid; i < n; i += 256) v[i] = g[i] * (v[i] - m) * inv + be[i];
    __syncthreads();
}

// ---------------------------------------------------------------------------
// Kernel 4: recurrent attention scan. One workgroup per batch row (64 WGs),
// each runs the 80-step serial loop out of LDS (latency-critical path).
// enc: (64,214,80) -> y: (64,80,80)
// ---------------------------------------------------------------------------
__global__ __launch_bounds__(256) void scan_kernel(
    const float* __restrict__ enc, const float* __restrict__ initQ,
    const float* __restrict__ q1w, const float* __restrict__ q1b,
    const float* __restrict__ g1,  const float* __restrict__ be1,
    const float* __restrict__ q2w, const float* __restrict__ q2b,
    const float* __restrict__ g2,  const float* __restrict__ be2,
    float* __restrict__ y)
{
    const int b = blockIdx.x;
    const int tid = threadIdx.x;
    const float* E = enc + b * (SEQ * DIMS);
    const float scale = 0.11180339887498948f;  // 1/sqrt(80)

    __shared__ float sp[80];     // prev / out
    __shared__ float sh[160];    // hidden
    __shared__ float sq[80];     // query
    __shared__ float spr[SEQ];   // scores -> probs
    __shared__ float red[256];

    // initial attention with initQ
    for (int s = tid; s < SEQ; s += 256) {
        float d = 0.f;
        for (int k = 0; k < 80; ++k) d = fmaf(initQ[k], E[s * 80 + k], d);
        spr[s] = d * scale;
    }
    __syncthreads();
    softmax_shared(spr, SEQ, red);
    for (int d = tid; d < 80; d += 256) {
        float a = 0.f;
        for (int s = 0; s < SEQ; ++s) a = fmaf(spr[s], E[s * 80 + d], a);
        sp[d] = a;
        y[(b * MAXL2 + 0) * 80 + d] = a;
    }
    __syncthreads();

    for (int t = 1; t < MAXL2; ++t) {
        // hq = relu(prev @ q1_w^T + q1_b); LN
        if (tid < 160) {
            float a = q1b[tid];
            for (int k = 0; k < 80; ++k) a = fmaf(sp[k], q1w[tid * 80 + k], a);
            sh[tid] = fmaxf(a, 0.f);
        }
        __syncthreads();
        layernorm_shared(sh, 160, g1, be1, red, 159.f);
        // q = LN(prev + hq @ q2_w^T + q2_b)
        if (tid < 80) {
            float a = q2b[tid] + sp[tid];
            for (int k = 0; k < 160; ++k) a = fmaf(sh[k], q2w[tid * 160 + k], a);
            sq[tid] = a;
        }
        __syncthreads();
        layernorm_shared(sq, 80, g2, be2, red, 79.f);
        // attention
        for (int s = tid; s < SEQ; s += 256) {
            float d = 0.f;
            for (int k = 0; k < 80; ++k) d = fmaf(sq[k], E[s * 80 + k], d);
            spr[s] = d * scale;
        }
        __syncthreads();
        softmax_shared(spr, SEQ, red);
        for (int d = tid; d < 80; d += 256) {
            float a = 0.f;
            for (int s = 0; s < SEQ; ++s) a = fmaf(spr[s], E[s * 80 + d], a);
            sp[d] = a;
            y[(b * MAXL2 + t) * 80 + d] = a;
        }
        __syncthreads();
    }
}

// ---------------------------------------------------------------------------
// Kernel 5: decoder: relu(y@W3^T+b3) -> LN -> @W4^T+b4 -> log_softmax
// y (5120,80) -> out (5120,95). 80 blocks x 128 threads, fp32 WMMA.
// ---------------------------------------------------------------------------
__global__ __launch_bounds__(128) void decoder_kernel(
    const float* __restrict__ y,
    const float* __restrict__ w3, const float* __restrict__ b3,
    const float* __restrict__ g3, const float* __restrict__ be3,
    const float* __restrict__ w4, const float* __restrict__ b4,
    float* __restrict__ out)
{
    __shared__ float sX[64 * 96];    // input rows (stride 80), then logits (stride 96)
    __shared__ float sT[64 * 128];

    const int tid  = threadIdx.x;
    const int wid  = tid >> 5;
    const int lane = tid & 31;
    const int r    = lane & 15;
    const int koff = (lane >> 4) << 1;
    const int mloc = (lane >> 4) << 3;
    const int row0 = blockIdx.x * 64;

    for (int i = tid; i < 64 * 80; i += 128) sX[i] = y[row0 * 80 + i];
    __syncthreads();

    // GEMM1: (64x80)@(80x128)
    v8f acc[8];
#pragma unroll
    for (int nt = 0; nt < 8; ++nt)
#pragma unroll
        for (int e = 0; e < 8; ++e) acc[nt][e] = 0.f;

    const float* aRow = &sX[(wid * 16 + r) * 80];
    for (int kb = 0; kb < 80; kb += 4) {
        v2f a; a[0] = aRow[kb + koff]; a[1] = aRow[kb + koff + 1];
#pragma unroll
        for (int nt = 0; nt < 8; ++nt) {
            int n = nt * 16 + r;
            v2f bf; bf[0] = w3[n * 80 + kb + koff]; bf[1] = w3[n * 80 + kb + koff + 1];
            acc[nt] = __builtin_amdgcn_wmma_f32_16x16x4_f32(
                false, a, false, bf, (short)0, acc[nt], false, false);
        }
    }
#pragma unroll
    for (int nt = 0; nt < 8; ++nt) {
        int n = nt * 16 + r;
        float bias = b3[n];
#pragma unroll
        for (int v = 0; v < 8; ++v)
            sT[(wid * 16 + mloc + v) * 128 + n] = fmaxf(acc[nt][v] + bias, 0.f);
    }
    __syncthreads();

    // LN over 128
    if (tid < 64) {
        float m = 0.f;
        for (int k = 0; k < 128; ++k) m += sT[tid * 128 + k];
        m *= (1.f / 128.f);
        float v = 0.f;
        for (int k = 0; k < 128; ++k) { float d = sT[tid * 128 + k] - m; v = fmaf(d, d, v); }
        float inv = 1.f / (sqrtf(v / 127.f) + 1e-6f);
        for (int k = 0; k < 128; ++k)
            sT[tid * 128 + k] = g3[k] * (sT[tid * 128 + k] - m) * inv + be3[k];
    }
    __syncthreads();

    // GEMM2: (64x128)@(128x95), 6 N-tiles (pad to 96), guard n<95
    v8f acc2[6];
#pragma unroll
    for (int nt = 0; nt < 6; ++nt)
#pragma unroll
        for (int e = 0; e < 8; ++e) acc2[nt][e] = 0.f;

    const float* aRow2 = &sT[(wid * 16 + r) * 128];
    for (int kb = 0; kb < 128; kb += 4) {
        v2f a; a[0] = aRow2[kb + koff]; a[1] = aRow2[kb + koff + 1];
#pragma unroll
        for (int nt = 0; nt < 6; ++nt) {
            int n = nt * 16 + r;
            v2f bf;
            bf[0] = (n < VOCAB) ? w4[n * 128 + kb + koff]     : 0.f;
            bf[1] = (n < VOCAB) ? w4[n * 128 + kb + koff + 1] : 0.f;
            acc2[nt] = __builtin_amdgcn_wmma_f32_16x16x4_f32(
                false, a, false, bf, (short)0, acc2[nt], false, false);
        }
    }
#pragma unroll
    for (int nt = 0; nt < 6; ++nt) {
        int n = nt * 16 + r;
        if (n < VOCAB) {
            float bias = b4[n];
#pragma unroll
            for (int v = 0; v < 8; ++v)
                sX[(wid * 16 + mloc + v) * 96 + n] = acc2[nt][v] + bias;
        }
    }
    __syncthreads();

    // log_softmax over 95, write out
    if (tid < 64) {
        float mx = -INFINITY;
        for (int k = 0; k < VOCAB; ++k) mx = fmaxf(mx, sX[tid * 96 + k]);
        float sum = 0.f;
        for (int k = 0; k < VOCAB; ++k) sum += expf(sX[tid * 96 + k] - mx);
        float lse = mx + logf(sum);
        for (int k = 0; k < VOCAB; ++k)
            out[(row0 + tid) * VOCAB + k] = sX[tid * 96 + k] - lse;
    }
}

// ---------------------------------------------------------------------------
extern "C" void kernel_launch(void* const* d_in, const int* in_sizes, int n_in,
                              void* d_out, int out_size, void* d_ws, size_t ws_size,
                              hipStream_t stream)
{
    (void)in_sizes; (void)n_in; (void)out_size; (void)ws_size;
    const float* x    = (const float*)d_in[0];
    const float* c1w  = (const float*)d_in[1];
    const float* c1b  = (const float*)d_in[2];
    const float* c2w  = (const float*)d_in[3];
    const float* c2b  = (const float*)d_in[4];
    const float* d1w  = (const float*)d_in[5];
    const float* d1b  = (const float*)d_in[6];
    const float* n1w  = (const float*)d_in[7];
    const float* n1b  = (const float*)d_in[8];
    const float* d2w  = (const float*)d_in[9];
    const float* d2b  = (const float*)d_in[10];
    const float* n2w  = (const float*)d_in[11];
    const float* n2b  = (const float*)d_in[12];
    const float* initQ= (const float*)d_in[13];
    const float* q1w  = (const float*)d_in[14];
    const float* q1b  = (const float*)d_in[15];
    const float* nq1w = (const float*)d_in[16];
    const float* nq1b = (const float*)d_in[17];
    const float* q2w  = (const float*)d_in[18];
    const float* q2b  = (const float*)d_in[19];
    const float* nq2w = (const float*)d_in[20];
    const float* nq2b = (const float*)d_in[21];
    const float* d3w  = (const float*)d_in[22];
    const float* d3b  = (const float*)d_in[23];
    const float* n3w  = (const float*)d_in[24];
    const float* n3b  = (const float*)d_in[25];
    const float* d4w  = (const float*)d_in[26];
    const float* d4b  = (const float*)d_in[27];
    float* out = (float*)d_out;

    // scratch layout (floats): h1 | enc0 | enc | y   (~21 MB total)
    float* h1   = (float*)d_ws;
    float* enc0 = h1   + (size_t)NB * 8 * 12 * 428;    // 2,629,632
    float* enc  = enc0 + (size_t)NB * SEQ * DIMS;      // 1,095,680
    float* yy   = enc  + (size_t)NB * SEQ * DIMS;      // 1,095,680

    {
        int total = NB * 8 * 12 * 428;
        conv1_pool_kernel<<<(total + 255) / 256, 256, 0, stream>>>(x, c1w, c1b, h1);
    }
    {
        int total = NB * 16 * 5 * SEQ;
        conv2_pool_kernel<<<(total + 255) / 256, 256, 0, stream>>>(h1, c2w, c2b, enc0);
    }
    encoder_mlp_kernel<<<SEQ, 128, 0, stream>>>(enc0, d1w, d1b, n1w, n1b,
                                                d2w, d2b, n2w, n2b, enc);
    scan_kernel<<<NB, 256, 0, stream>>>(enc, initQ, q1w, q1b, nq1w, nq1b,
                                        q2w, q2b, nq2w, nq2b, yy);
    decoder_kernel<<<MAXL2, 128, 0, stream>>>(yy, d3w, d3b, n3w, n3b, d4w, d4b, out);
}